// TTTBlock_2929167696274
// MI455X (gfx1250) — compile-verified
//
#include <hip/hip_runtime.h>
#include <hip/hip_bf16.h>
#include <math.h>

// ---------------- problem constants (match reference) ----------------
constexpr int kB   = 16;
constexpr int kH   = 56;
constexpr int kW   = 56;
constexpr int kC   = 192;
constexpr int kNH  = 6;
constexpr int kL   = kH * kW;        // 3136
constexpr int kDH  = 32;
constexpr int kHID = 768;
constexpr int kM   = kB * kL;        // 50176 rows
constexpr float kETA = 1.0f / 32.0f;
constexpr float kEPS = 1e-5f;

// TTT chunking
constexpr int S    = 16;             // chunk length
constexpr int NCH  = kL / S;         // 196 chunks
constexpr int PRE_R  = 0;            // R = K^T T   (32x16)
constexpr int PRE_U  = 512;          // U = V^T T   (32x16)
constexpr int PRE_M  = 1024;         // Mm = -eta*mask*(K Q^T) (16x16)
constexpr int PRE_SZ = 1280;         // floats per (head,chunk)

typedef __attribute__((ext_vector_type(2))) float v2f;
typedef __attribute__((ext_vector_type(8))) float v8f;

__device__ __forceinline__ v8f wmma4(v2f a, v2f b, v8f c) {
    // D(16x16,f32) = A(16x4,f32) * B(4x16,f32) + C
    return __builtin_amdgcn_wmma_f32_16x16x4_f32(false, a, false, b, (short)0, c,
                                                 false, false);
}

__device__ __forceinline__ float gelu_exact(float x) {
    return 0.5f * x * (1.0f + erff(x * 0.70710678118654752f));
}

// =====================================================================
// Depthwise 3x3 conv residual:  y = f(x + dwconv(x,w) + b), f = id|gelu
// =====================================================================
template <int CH, bool GELU_OUT>
__global__ __launch_bounds__(256) void dwconv_kernel(
    const float* __restrict__ x, const float* __restrict__ w,
    const float* __restrict__ bias, float* __restrict__ y)
{
    size_t idx = (size_t)blockIdx.x * 256 + threadIdx.x;
    if (idx >= (size_t)kM * CH) return;
    int    c  = (int)(idx % CH);
    size_t bl = idx / CH;
    int    l  = (int)(bl % kL);
    size_t bb = bl / kL;
    int hh = l / kW, ww = l % kW;

    float acc = bias[c];
#pragma unroll
    for (int dy = 0; dy < 3; dy++) {
        int yy = hh + dy - 1;
        if (yy < 0 || yy >= kH) continue;
#pragma unroll
        for (int dx = 0; dx < 3; dx++) {
            int xx = ww + dx - 1;
            if (xx < 0 || xx >= kW) continue;
            acc += w[c * 9 + dy * 3 + dx] *
                   x[(bb * kL + (size_t)(yy * kW + xx)) * CH + c];
        }
    }
    float v = x[idx] + acc;
    if (GELU_OUT) v = gelu_exact(v);
    y[idx] = v;
}

// =====================================================================
// LayerNorm over C=192: one wave32 per row, 6 elems/lane, shfl reduce
// =====================================================================
__global__ __launch_bounds__(256) void ln_kernel(
    const float* __restrict__ x, const float* __restrict__ g,
    const float* __restrict__ b, float* __restrict__ y, int rows)
{
    int wv = threadIdx.x >> 5, lane = threadIdx.x & 31;
    int row = blockIdx.x * 8 + wv;
    if (row >= rows) return;
    const float* xr = x + (size_t)row * kC;
    float v[6], s = 0.f;
#pragma unroll
    for (int i = 0; i < 6; i++) { v[i] = xr[lane + 32 * i]; s += v[i]; }
#pragma unroll
    for (int off = 16; off; off >>= 1) s += __shfl_xor(s, off, 32);
    float mu = s * (1.0f / kC);
    float q = 0.f;
#pragma unroll
    for (int i = 0; i < 6; i++) { float d = v[i] - mu; q += d * d; }
#pragma unroll
    for (int off = 16; off; off >>= 1) q += __shfl_xor(q, off, 32);
    float rstd = rsqrtf(q * (1.0f / kC) + kEPS);
    float* yr = y + (size_t)row * kC;
#pragma unroll
    for (int i = 0; i < 6; i++) {
        int c = lane + 32 * i;
        yr[c] = (v[i] - mu) * rstd * g[c] + b[c];
    }
}

// =====================================================================
// RoPE on q,k sections of qkv (in place); q also scaled by DH^-0.5
// =====================================================================
__global__ __launch_bounds__(256) void rope_kernel(float* __restrict__ qkv)
{
    size_t idx = (size_t)blockIdx.x * 256 + threadIdx.x;   // over M * 2 * (C/2)
    size_t total = (size_t)kM * kC;
    if (idx >= total) return;
    constexpr int kmax = kC / 4;                           // 48
    int    c2  = (int)(idx % (kC / 2));
    size_t r1  = idx / (kC / 2);
    int    sec = (int)(r1 % 2);                            // 0=q, 1=k
    size_t bl  = r1 / 2;
    int l = (int)(bl % kL);
    int hh = l / kW, ww = l % kW;

    float kidx  = (c2 < kmax) ? (float)c2 : (float)(c2 - kmax);
    float theta = expf(-(kidx / (float)kmax) * 9.210340371976184f); // ln(1e4)
    float ang   = ((c2 < kmax) ? (float)hh : (float)ww) * theta;
    float sn, cs;
    sincosf(ang, &sn, &cs);

    float* p = qkv + bl * (3 * kC) + (size_t)sec * kC + 2 * c2;
    float re = p[0], im = p[1];
    float o_re = cs * re - sn * im;
    float o_im = sn * re + cs * im;
    float scale = (sec == 0) ? 0.1767766952966369f : 1.0f;  // 32^-0.5 on q
    p[0] = o_re * scale;
    p[1] = o_im * scale;
}

// =====================================================================
// WMMA f32 GEMM, double-buffered LDS staging:
//   C[M,N] = epi(A[M,K] @ Wt[N,K]^T + bias [+ resid])
// block = 256 thr (8 waves), tile 128x64, BK=32, wave owns 32x32.
// EPI: 0 = bias, 1 = bias+gelu, 2 = bias+residual
// =====================================================================
template <int EPI>
__global__ __launch_bounds__(256) void gemm_kernel(
    const float* __restrict__ A, const float* __restrict__ Wt,
    const float* __restrict__ bias, const float* __restrict__ resid,
    float* __restrict__ Cout, int K, int N)
{
    constexpr int BM = 128, BN = 64, BK = 32, AST = BK + 4;  // pad vs bank conflicts
    __shared__ float Ald[2][BM * AST];
    __shared__ float Bld[2][BK * BN];

    const int m0 = blockIdx.x * BM;
    const int n0 = blockIdx.y * BN;
    const int tid  = threadIdx.x;
    const int lane = tid & 31;
    const int wv   = tid >> 5;
    const int wm   = (wv & 3) * 32;   // wave m-offset in block tile
    const int wn   = (wv >> 2) * 32;  // wave n-offset in block tile
    const int lh   = lane >> 4;       // half-wave select (0/1)
    const int ll   = lane & 15;

    const v8f vzero = {0.f, 0.f, 0.f, 0.f, 0.f, 0.f, 0.f, 0.f};
    v8f acc[2][2];
#pragma unroll
    for (int i = 0; i < 2; i++)
#pragma unroll
        for (int j = 0; j < 2; j++) acc[i][j] = vzero;

    const int nIter = K / BK;

    // ---- stage tile 0 ----
#pragma unroll
    for (int i = 0; i < 4; i++) {
        int idx = tid + 256 * i;               // 0..1023 float4s
        int row = idx >> 3;
        int c4  = (idx & 7) * 4;
        const float4 va =
            *reinterpret_cast<const float4*>(&A[(size_t)(m0 + row) * K + c4]);
        float* dst = &Ald[0][row * AST + c4];
        dst[0] = va.x; dst[1] = va.y; dst[2] = va.z; dst[3] = va.w;
    }
#pragma unroll
    for (int i = 0; i < 8; i++) {
        int idx = tid + 256 * i;               // 0..2047
        int n  = idx >> 5;
        int kk = idx & 31;
        Bld[0][kk * BN + n] = Wt[(size_t)(n0 + n) * K + kk];
    }
    __syncthreads();

    int cur = 0;
    for (int it = 0; it < nIter; ++it) {
        const int k0n = (it + 1) * BK;
        float4 ra[4];
        float  rb[8];
        if (it + 1 < nIter) {
            // issue next tile's global loads (no wait yet — overlap with WMMA)
#pragma unroll
            for (int i = 0; i < 4; i++) {
                int idx = tid + 256 * i;
                int row = idx >> 3;
                int c4  = (idx & 7) * 4;
                ra[i] = *reinterpret_cast<const float4*>(
                    &A[(size_t)(m0 + row) * K + k0n + c4]);
            }
#pragma unroll
            for (int i = 0; i < 8; i++) {
                int idx = tid + 256 * i;
                int n  = idx >> 5;
                int kk = idx & 31;
                rb[i] = Wt[(size_t)(n0 + n) * K + k0n + kk];
            }
            if (it + 2 < nIter)   // L2 prefetch the tile after next
                __builtin_prefetch(&A[(size_t)(m0 + (tid >> 1)) * K + k0n + BK],
                                   0, 0);
        }

        // ---- compute on current buffer ----
        const float* Ac = Ald[cur];
        const float* Bc = Bld[cur];
#pragma unroll
        for (int k4 = 0; k4 < 8; k4++) {
            const int kk = k4 * 4 + lh * 2;
            v2f afr[2], bfr[2];
#pragma unroll
            for (int mi = 0; mi < 2; mi++) {
                const float* ap = &Ac[(wm + mi * 16 + ll) * AST + kk];
                afr[mi] = (v2f){ap[0], ap[1]};
            }
#pragma unroll
            for (int ni = 0; ni < 2; ni++) {
                const float* bp = &Bc[kk * BN + wn + ni * 16 + ll];
                bfr[ni] = (v2f){bp[0], bp[BN]};
            }
#pragma unroll
            for (int mi = 0; mi < 2; mi++)
#pragma unroll
                for (int ni = 0; ni < 2; ni++)
                    acc[mi][ni] = wmma4(afr[mi], bfr[ni], acc[mi][ni]);
        }

        if (it + 1 < nIter) {
            float* An = Ald[cur ^ 1];
            float* Bn = Bld[cur ^ 1];
#pragma unroll
            for (int i = 0; i < 4; i++) {
                int idx = tid + 256 * i;
                int row = idx >> 3;
                int c4  = (idx & 7) * 4;
                float* dst = &An[row * AST + c4];
                dst[0] = ra[i].x; dst[1] = ra[i].y;
                dst[2] = ra[i].z; dst[3] = ra[i].w;
            }
#pragma unroll
            for (int i = 0; i < 8; i++) {
                int idx = tid + 256 * i;
                int n  = idx >> 5;
                int kk = idx & 31;
                Bn[kk * BN + n] = rb[i];
            }
        }
        __syncthreads();
        cur ^= 1;
    }

    // ---- epilogue ----
#pragma unroll
    for (int mi = 0; mi < 2; mi++)
#pragma unroll
        for (int ni = 0; ni < 2; ni++) {
#pragma unroll
            for (int r = 0; r < 8; r++) {
                int m = m0 + wm + mi * 16 + r + lh * 8;
                int n = n0 + wn + ni * 16 + ll;
                float v = acc[mi][ni][r] + bias[n];
                if constexpr (EPI == 1) v = gelu_exact(v);
                if constexpr (EPI == 2) v += resid[(size_t)m * N + n];
                Cout[(size_t)m * N + n] = v;
            }
        }
}

// =====================================================================
// TTT precompute (fully parallel over 96 heads x 196 chunks):
//   Gkk = K K^T ;  T = inv(I + eta*strictupper(Gkk))  (unit upper tri)
//   R = K^T T ; U = V^T T ; Mm = -eta * mask(s<=t) * (K Q^T)
// This removes the triangular solve from the serial scan entirely.
// =====================================================================
__global__ __launch_bounds__(32) void ttt_prep_kernel(
    const float* __restrict__ qkv, float* __restrict__ pre)
{
    constexpr int TST = 36;
    __shared__ float Kt[S * TST], Qt[S * TST], Vt[S * TST];  // [t][dh]
    __shared__ float Gkk[16 * 18];
    __shared__ float Tl[16 * 18];

    const int lane  = threadIdx.x;
    const int bh    = blockIdx.x;    // 0..B*NH-1
    const int c     = blockIdx.y;    // chunk
    const int bb    = bh / kNH;
    const int hh    = bh % kNH;
    const int hbase = hh * kDH;
    const int l0    = c * S;
    const int lh = lane >> 4, ll = lane & 15;
    const v8f vzero = {0.f, 0.f, 0.f, 0.f, 0.f, 0.f, 0.f, 0.f};

    for (int t = 0; t < S; t++) {
        size_t base = ((size_t)bb * kL + l0 + t) * (3 * kC) + hbase;
        Qt[t * TST + lane] = qkv[base + lane];
        Kt[t * TST + lane] = qkv[base + kC + lane];
        Vt[t * TST + lane] = qkv[base + 2 * kC + lane];
    }
    for (int i = lane; i < 16 * 18; i += 32) Tl[i] = 0.f;
    __syncthreads();

    float* preb = pre + ((size_t)bh * NCH + c) * PRE_SZ;

    // Gkk = K K^T
    {
        v8f acc = vzero;
#pragma unroll
        for (int k4 = 0; k4 < 8; k4++) {
            const int kk = k4 * 4 + lh * 2;
            v2f a = (v2f){Kt[ll * TST + kk], Kt[ll * TST + kk + 1]};
            acc = wmma4(a, a, acc);          // B frag of K^T == A frag of K
        }
#pragma unroll
        for (int r = 0; r < 8; r++)
            Gkk[(r + lh * 8) * 18 + ll] = acc[r];
    }
    // Mm = -eta * mask(s<=t) * (K Q^T)  -> straight to global
    {
        v8f acc = vzero;
#pragma unroll
        for (int k4 = 0; k4 < 8; k4++) {
            const int kk = k4 * 4 + lh * 2;
            v2f a = (v2f){Kt[ll * TST + kk], Kt[ll * TST + kk + 1]};
            v2f b = (v2f){Qt[ll * TST + kk], Qt[ll * TST + kk + 1]};
            acc = wmma4(a, b, acc);
        }
#pragma unroll
        for (int r = 0; r < 8; r++) {
            int s = r + lh * 8, t = ll;
            preb[PRE_M + s * 16 + t] = (s <= t) ? (-kETA) * acc[r] : 0.f;
        }
    }
    __syncthreads();

    // T = inv(I + eta*strictupper(Gkk)); lane t owns column t (back-subst)
    if (lane < 16) {
        const int t = lane;
        Tl[t * 18 + t] = 1.f;
        for (int s = t - 1; s >= 0; s--) {
            float a = 0.f;
            for (int r = s + 1; r <= t; r++)
                a += Gkk[s * 18 + r] * Tl[r * 18 + t];
            Tl[s * 18 + t] = -kETA * a;
        }
    }
    __syncthreads();

    // R = K^T T ; U = V^T T  (both 32x16, k-dim 16)
    for (int mo = 0; mo < 32; mo += 16) {
        v8f accR = vzero, accU = vzero;
#pragma unroll
        for (int k4 = 0; k4 < 4; k4++) {
            const int kk = k4 * 4 + lh * 2;
            v2f b  = (v2f){Tl[kk * 18 + ll], Tl[(kk + 1) * 18 + ll]};
            v2f aK = (v2f){Kt[kk * TST + mo + ll], Kt[(kk + 1) * TST + mo + ll]};
            v2f aV = (v2f){Vt[kk * TST + mo + ll], Vt[(kk + 1) * TST + mo + ll]};
            accR = wmma4(aK, b, accR);
            accU = wmma4(aV, b, accU);
        }
#pragma unroll
        for (int r = 0; r < 8; r++) {
            int e = mo + r + lh * 8;
            preb[PRE_R + e * 16 + ll] = accR[r];
            preb[PRE_U + e * 16 + ll] = accU[r];
        }
    }
}

// =====================================================================
// TTT serial scan (one wave32 per (batch,head)); critical path is now
// pure chained WMMA:  G = W R - U ;  O = W Q^T + G Mm ;  W -= eta G K
// =====================================================================
__global__ __launch_bounds__(32) void ttt_scan_kernel(
    const float* __restrict__ qkv, const float* __restrict__ W0,
    const float* __restrict__ pre, float* __restrict__ attn)
{
    constexpr int TST = 36, WST = 36, GST = 18;
    __shared__ float Wl[32 * WST];           // live W state
    __shared__ float Kt[S * TST], Qt[S * TST];
    __shared__ float Rl[32 * 16], Ul[32 * 16], Ml[16 * 16];
    __shared__ float Gl[32 * GST];           // G  [d][t]

    const int lane  = threadIdx.x;
    const int bh    = blockIdx.x;
    const int bb    = bh / kNH;
    const int hh    = bh % kNH;
    const int hbase = hh * kDH;
    const int lh = lane >> 4, ll = lane & 15;
    const v8f vzero = {0.f, 0.f, 0.f, 0.f, 0.f, 0.f, 0.f, 0.f};

    for (int r = 0; r < 32; r++)
        Wl[r * WST + lane] = W0[(hh * 32 + r) * 32 + lane];
    __syncthreads();

    for (int c = 0; c < NCH; c++) {
        const int l0 = c * S;
        const float* preb = pre + ((size_t)bh * NCH + c) * PRE_SZ;

        // prefetch next chunk's precompute block into L2
        if (c + 1 < NCH) {
            const float* pn = preb + PRE_SZ;
            for (int i = lane * 16; i < PRE_SZ; i += 32 * 16)
                __builtin_prefetch(&pn[i], 0, 0);
        }

        for (int t = 0; t < S; t++) {
            size_t base = ((size_t)bb * kL + l0 + t) * (3 * kC) + hbase;
            Qt[t * TST + lane] = qkv[base + lane];
            Kt[t * TST + lane] = qkv[base + kC + lane];
        }
        for (int i = lane; i < 512; i += 32) {
            Rl[i] = preb[PRE_R + i];
            Ul[i] = preb[PRE_U + i];
        }
        for (int i = lane; i < 256; i += 32) Ml[i] = preb[PRE_M + i];
        __syncthreads();

        // ---- G = W R - U  (32x16) ----
        for (int mo = 0; mo < 32; mo += 16) {
            v8f acc;
#pragma unroll
            for (int r = 0; r < 8; r++)
                acc[r] = -Ul[(mo + r + lh * 8) * 16 + ll];
#pragma unroll
            for (int k4 = 0; k4 < 8; k4++) {
                const int kk = k4 * 4 + lh * 2;
                const float* ap = &Wl[(mo + ll) * WST + kk];
                v2f a = (v2f){ap[0], ap[1]};
                v2f b = (v2f){Rl[kk * 16 + ll], Rl[(kk + 1) * 16 + ll]};
                acc = wmma4(a, b, acc);
            }
#pragma unroll
            for (int r = 0; r < 8; r++)
                Gl[(mo + r + lh * 8) * GST + ll] = acc[r];
        }
        __syncthreads();

        // ---- O = W Q^T + G Mm ; store to attn (B,L,C) ----
        for (int mo = 0; mo < 32; mo += 16) {
            v8f acc = vzero;
#pragma unroll
            for (int k4 = 0; k4 < 8; k4++) {
                const int kk = k4 * 4 + lh * 2;
                const float* ap = &Wl[(mo + ll) * WST + kk];
                v2f a = (v2f){ap[0], ap[1]};
                v2f b = (v2f){Qt[ll * TST + kk], Qt[ll * TST + kk + 1]};
                acc = wmma4(a, b, acc);
            }
#pragma unroll
            for (int k4 = 0; k4 < 4; k4++) {
                const int kk = k4 * 4 + lh * 2;
                const float* gp = &Gl[(mo + ll) * GST + kk];
                v2f a = (v2f){gp[0], gp[1]};
                v2f b = (v2f){Ml[kk * 16 + ll], Ml[(kk + 1) * 16 + ll]};
                acc = wmma4(a, b, acc);
            }
#pragma unroll
            for (int r = 0; r < 8; r++) {
                int d = mo + r + lh * 8;
                attn[((size_t)bb * kL + l0 + ll) * kC + hbase + d] = acc[r];
            }
        }

        // ---- W <- W - eta * G K  (32x32) ----
        for (int mo = 0; mo < 32; mo += 16)
            for (int no = 0; no < 32; no += 16) {
                v8f acc;
#pragma unroll
                for (int r = 0; r < 8; r++)
                    acc[r] = Wl[(mo + r + lh * 8) * WST + no + ll];
#pragma unroll
                for (int k4 = 0; k4 < 4; k4++) {
                    const int kk = k4 * 4 + lh * 2;
                    const float* gp = &Gl[(mo + ll) * GST + kk];
                    v2f a = (v2f){gp[0], gp[1]};
                    v2f b = (v2f){-kETA * Kt[kk * TST + no + ll],
                                  -kETA * Kt[(kk + 1) * TST + no + ll]};
                    acc = wmma4(a, b, acc);
                }
#pragma unroll
                for (int r = 0; r < 8; r++)
                    Wl[(mo + r + lh * 8) * WST + no + ll] = acc[r];
            }
        __syncthreads();
    }
}

// =====================================================================
// launch
// =====================================================================
extern "C" void kernel_launch(void* const* d_in, const int* in_sizes, int n_in,
                              void* d_out, int out_size, void* d_ws, size_t ws_size,
                              hipStream_t stream)
{
    (void)in_sizes; (void)n_in; (void)out_size; (void)ws_size;

    const float* x      = (const float*)d_in[0];
    const float* cpe_w  = (const float*)d_in[1];
    const float* cpe_b  = (const float*)d_in[2];
    const float* n1_g   = (const float*)d_in[3];
    const float* n1_b   = (const float*)d_in[4];
    const float* qkv_w  = (const float*)d_in[5];
    const float* qkv_b  = (const float*)d_in[6];
    const float* W0     = (const float*)d_in[7];
    const float* proj_w = (const float*)d_in[8];
    const float* proj_b = (const float*)d_in[9];
    const float* n2_g   = (const float*)d_in[10];
    const float* n2_b   = (const float*)d_in[11];
    const float* fc1_w  = (const float*)d_in[12];
    const float* fc1_b  = (const float*)d_in[13];
    const float* dwc_w  = (const float*)d_in[14];
    const float* dwc_b  = (const float*)d_in[15];
    const float* fc2_w  = (const float*)d_in[16];
    const float* fc2_b  = (const float*)d_in[17];
    float* out = (float*)d_out;

    float* ws  = (float*)d_ws;
    float* X1  = ws;                           // (M,C)  x after CPE
    float* XN  = X1  + (size_t)kM * kC;        // (M,C)  ln1 out -> reused as attn
    float* QKV = XN  + (size_t)kM * kC;        // (M,3C)
    float* X2  = QKV + (size_t)kM * 3 * kC;    // (M,C)  post-attn residual
    float* H1  = X2  + (size_t)kM * kC;        // (M,HID)
    float* H2  = H1  + (size_t)kM * kHID;      // (M,HID)
    float* PRE = H2  + (size_t)kM * kHID;      // (96*196*1280) TTT precompute

    // 1. CPE: X1 = x + dwconv(x)
    dwconv_kernel<kC, false>
        <<<(unsigned)(((size_t)kM * kC + 255) / 256), 256, 0, stream>>>(
            x, cpe_w, cpe_b, X1);
    // 2. LN1: XN = layernorm(X1)
    ln_kernel<<<kM / 8, 256, 0, stream>>>(X1, n1_g, n1_b, XN, kM);
    // 3. QKV = XN @ qkv_w^T + b
    gemm_kernel<0><<<dim3(kM / 128, (3 * kC) / 64), 256, 0, stream>>>(
        XN, qkv_w, qkv_b, nullptr, QKV, kC, 3 * kC);
    // 4. RoPE (+ q scale) in place on QKV
    rope_kernel<<<(unsigned)(((size_t)kM * kC + 255) / 256), 256, 0, stream>>>(QKV);
    // 5a. TTT precompute: all chunks in parallel
    ttt_prep_kernel<<<dim3(kB * kNH, NCH), 32, 0, stream>>>(QKV, PRE);
    // 5b. TTT serial scan: attn -> XN (reuse)
    ttt_scan_kernel<<<kB * kNH, 32, 0, stream>>>(QKV, W0, PRE, XN);
    // 6. X2 = X1 + attn @ proj_w^T + b
    gemm_kernel<2><<<dim3(kM / 128, kC / 64), 256, 0, stream>>>(
        XN, proj_w, proj_b, X1, X2, kC, kC);
    // 7. LN2: X1 = layernorm(X2)  (X1 reused)
    ln_kernel<<<kM / 8, 256, 0, stream>>>(X2, n2_g, n2_b, X1, kM);
    // 8. H1 = gelu(X1 @ fc1_w^T + b)
    gemm_kernel<1><<<dim3(kM / 128, kHID / 64), 256, 0, stream>>>(
        X1, fc1_w, fc1_b, nullptr, H1, kC, kHID);
    // 9. H2 = gelu(H1 + dwconv(H1))
    dwconv_kernel<kHID, true>
        <<<(unsigned)(((size_t)kM * kHID + 255) / 256), 256, 0, stream>>>(
            H1, dwc_w, dwc_b, H2);
    // 10. out = X2 + H2 @ fc2_w^T + b
    gemm_kernel<2><<<dim3(kM / 128, kC / 64), 256, 0, stream>>>(
        H2, fc2_w, fc2_b, X2, out, kHID, kC);
}